// CustomGATGraph_36567351558182
// MI455X (gfx1250) — compile-verified
//
#include <hip/hip_runtime.h>
#include <cstdint>
#include <cstddef>

// Problem constants (from the reference)
#define NN     50000                 // nodes
#define EE     800000                // raw edges
#define ETOT   (EE + NN)             // edges + self loops
#define DINF   128
#define HIDF   128                   // HEADS*DH
#define NHEADS 4
#define DHH    32
#define DOUTF  16
#define NLHID  2

typedef float v2f __attribute__((ext_vector_type(2)));
typedef float v8f __attribute__((ext_vector_type(8)));

// ---- order-preserving float<->uint encoding for atomicMax on floats ----
__device__ __forceinline__ unsigned enc_f32(float f) {
  unsigned u = __float_as_uint(f);
  return (u & 0x80000000u) ? ~u : (u | 0x80000000u);
}
__device__ __forceinline__ float dec_f32(unsigned u) {
  return (u & 0x80000000u) ? __uint_as_float(u ^ 0x80000000u)
                           : __uint_as_float(~u);
}
// enc(-inf) = ~0xFF800000 = 0x007FFFFF
#define ENC_NEG_INF 0x007FFFFFu

// =====================================================================
// GEMM: H[M,NOUT] = X[M,K] @ W[K,NOUT]  (fp32, V_WMMA_F32_16X16X4_F32)
// One wave (32 lanes) computes one 16x16 output tile.
// A 16x4 layout: lanes 0-15 -> M=0..15 {K=k0,k0+1}, lanes 16-31 -> {K=k0+2,k0+3}
// B 4x16 layout: symmetric (row striped across lanes within a VGPR).
// C/D 16x16:     vgpr r, lane l -> M = r + 8*(l/16), N = l%16
// =====================================================================
template<int K, int NOUT>
__global__ __launch_bounds__(32)
void gemm_f32_wmma(const float* __restrict__ X, const float* __restrict__ W,
                   float* __restrict__ H) {
  const int lane = threadIdx.x & 31;
  const int half = lane >> 4;     // 0 or 1
  const int l16  = lane & 15;
  const int row  = blockIdx.x * 16 + l16;   // A-matrix row for this lane
  const int col  = blockIdx.y * 16 + l16;   // B/D column for this lane
  v8f acc = {};
#pragma unroll
  for (int k0 = 0; k0 < K; k0 += 4) {
    const int ka = k0 + half * 2;
    v2f a, b;
    a.x = X[row * K + ka];
    a.y = X[row * K + ka + 1];
    b.x = W[ka * NOUT + col];
    b.y = W[(ka + 1) * NOUT + col];
    acc = __builtin_amdgcn_wmma_f32_16x16x4_f32(false, a, false, b,
                                                (short)0, acc, false, false);
  }
#pragma unroll
  for (int r = 0; r < 8; ++r) {
    const int m = blockIdx.x * 16 + r + 8 * half;
    H[m * NOUT + col] = acc[r];
  }
}

// =====================================================================
// Per-(node,head) attention logits: al_s = <h[n,h,:], a_s[h,:]>, same for a_d
// =====================================================================
__global__ __launch_bounds__(256)
void att_logits(const float* __restrict__ Hm, const float* __restrict__ a_s,
                const float* __restrict__ a_d, float* __restrict__ als,
                float* __restrict__ ald, int heads, int ch) {
  const int idx = blockIdx.x * blockDim.x + threadIdx.x;  // n*heads + h
  if (idx >= NN * heads) return;
  const int h = idx % heads;
  const float* hv = Hm + idx * ch;  // (n*heads + h)*ch
  float s = 0.f, d = 0.f;
  for (int c = 0; c < ch; ++c) {
    const float v = hv[c];
    s += v * a_s[h * ch + c];
    d += v * a_d[h * ch + c];
  }
  als[idx] = s;
  ald[idx] = d;
}

// ---- fills ----
__global__ __launch_bounds__(256)
void fill_u32(unsigned* __restrict__ p, unsigned v, int n) {
  const int i = blockIdx.x * blockDim.x + threadIdx.x;
  if (i < n) p[i] = v;
}
__global__ __launch_bounds__(256)
void fill_f32(float* __restrict__ p, float v, int n) {
  const int i = blockIdx.x * blockDim.x + threadIdx.x;
  if (i < n) p[i] = v;
}

__device__ __forceinline__ void edge_endpoints(const int* __restrict__ ei,
                                               int e, int& src, int& dst) {
  if (e < EE) { src = ei[e]; dst = ei[EE + e]; }
  else        { src = e - EE; dst = e - EE; }   // self loop
}

// =====================================================================
// Pass 1: e = leaky_relu(al_s[src] + al_d[dst]); scatter-max into dst
// =====================================================================
__global__ __launch_bounds__(256)
void edge_logit_max(const int* __restrict__ ei, const float* __restrict__ als,
                    const float* __restrict__ ald, unsigned* __restrict__ menc,
                    float* __restrict__ ebuf, int heads) {
  const int e = blockIdx.x * blockDim.x + threadIdx.x;
  if (e >= ETOT) return;
  int src, dst;
  edge_endpoints(ei, e, src, dst);
  for (int h = 0; h < heads; ++h) {
    const float x  = als[src * heads + h] + ald[dst * heads + h];
    const float lr = (x > 0.f) ? x : 0.2f * x;     // leaky_relu(0.2)
    ebuf[e * heads + h] = lr;
    atomicMax(&menc[dst * heads + h], enc_f32(lr));
  }
}

// =====================================================================
// Pass 2: ee = exp(e - max[dst]); scatter-add denom[dst]; ebuf <- ee
// =====================================================================
__global__ __launch_bounds__(256)
void edge_exp_sum(const int* __restrict__ ei, float* __restrict__ ebuf,
                  const unsigned* __restrict__ menc, float* __restrict__ denom,
                  int heads) {
  const int e = blockIdx.x * blockDim.x + threadIdx.x;
  if (e >= ETOT) return;
  int src, dst;
  edge_endpoints(ei, e, src, dst);
  for (int h = 0; h < heads; ++h) {
    const float m  = dec_f32(menc[dst * heads + h]);
    const float ee = __expf(ebuf[e * heads + h] - m);
    ebuf[e * heads + h] = ee;
    unsafeAtomicAdd(&denom[dst * heads + h], ee);
  }
}

// =====================================================================
// Pass 3: agg[dst, c] += h[src, c] * ee / (denom[dst] + eps)
// thread per (edge, channel)
// =====================================================================
__global__ __launch_bounds__(256)
void edge_scatter(const int* __restrict__ ei, const float* __restrict__ Hm,
                  const float* __restrict__ ebuf, const float* __restrict__ denom,
                  float* __restrict__ agg, int heads, int hid) {
  const int gid = blockIdx.x * blockDim.x + threadIdx.x;
  if (gid >= ETOT * hid) return;
  const int e = gid / hid;
  const int c = gid % hid;
  const int h = c / (hid / heads);
  int src, dst;
  edge_endpoints(ei, e, src, dst);
  const float alpha = ebuf[e * heads + h] / (denom[dst * heads + h] + 1e-16f);
  unsafeAtomicAdd(&agg[dst * hid + c], Hm[src * hid + c] * alpha);
}

// =====================================================================
// Finalize: out = agg + bias, optional ELU
// =====================================================================
__global__ __launch_bounds__(256)
void finalize(float* __restrict__ agg, const float* __restrict__ bias,
              int hid, int apply_elu) {
  const int idx = blockIdx.x * blockDim.x + threadIdx.x;
  if (idx >= NN * hid) return;
  float v = agg[idx] + bias[idx % hid];
  if (apply_elu) v = (v > 0.f) ? v : (__expf(v) - 1.f);
  agg[idx] = v;
}

// =====================================================================
// host side
// =====================================================================
static inline void* wsalloc(char* base, size_t& off, size_t bytes) {
  void* p = base + off;
  off += (bytes + 255) & ~(size_t)255;
  return p;
}

static void run_attention(const int* ei, const float* Hm, const float* a_s,
                          const float* a_d, const float* bias, float* agg,
                          float* als, float* ald, unsigned* menc, float* denom,
                          float* ebuf, int heads, int ch, int apply_elu,
                          hipStream_t stream) {
  const int hid = heads * ch;
  const int nh  = NN * heads;
  att_logits<<<(nh + 255) / 256, 256, 0, stream>>>(Hm, a_s, a_d, als, ald, heads, ch);
  fill_u32<<<(nh + 255) / 256, 256, 0, stream>>>(menc, ENC_NEG_INF, nh);
  fill_f32<<<(nh + 255) / 256, 256, 0, stream>>>(denom, 0.f, nh);
  fill_f32<<<(NN * hid + 255) / 256, 256, 0, stream>>>(agg, 0.f, NN * hid);
  edge_logit_max<<<(ETOT + 255) / 256, 256, 0, stream>>>(ei, als, ald, menc, ebuf, heads);
  edge_exp_sum<<<(ETOT + 255) / 256, 256, 0, stream>>>(ei, ebuf, menc, denom, heads);
  {
    const int total = ETOT * hid;
    edge_scatter<<<(total + 255) / 256, 256, 0, stream>>>(ei, Hm, ebuf, denom, agg, heads, hid);
  }
  finalize<<<(NN * hid + 255) / 256, 256, 0, stream>>>(agg, bias, hid, apply_elu);
}

extern "C" void kernel_launch(void* const* d_in, const int* in_sizes, int n_in,
                              void* d_out, int out_size, void* d_ws, size_t ws_size,
                              hipStream_t stream) {
  const float* x   = (const float*)d_in[0];
  const int*   ei  = (const int*)d_in[1];   // [2, E]: row0 = src, row1 = dst
  // d_in[2] = edge_attr (unused by the reference GATConv)
  const float* W1  = (const float*)d_in[3];
  const float* as1 = (const float*)d_in[4];
  const float* ad1 = (const float*)d_in[5];
  const float* b1  = (const float*)d_in[6];
  const float* Wh  = (const float*)d_in[7];   // [NLHID,128,128]
  const float* ash = (const float*)d_in[8];   // [NLHID,4,32]
  const float* adh = (const float*)d_in[9];
  const float* bh  = (const float*)d_in[10];  // [NLHID,128]
  const float* W2  = (const float*)d_in[11];  // [128,16]
  const float* as2 = (const float*)d_in[12];
  const float* ad2 = (const float*)d_in[13];
  const float* b2  = (const float*)d_in[14];
  float* out = (float*)d_out;                 // [N,16] fp32

  char* ws = (char*)d_ws;
  size_t off = 0;
  float*    h0    = (float*)   wsalloc(ws, off, (size_t)NN * HIDF * 4);  // gemm out
  float*    h1    = (float*)   wsalloc(ws, off, (size_t)NN * HIDF * 4);  // agg / next input
  float*    als   = (float*)   wsalloc(ws, off, (size_t)NN * NHEADS * 4);
  float*    ald   = (float*)   wsalloc(ws, off, (size_t)NN * NHEADS * 4);
  unsigned* menc  = (unsigned*)wsalloc(ws, off, (size_t)NN * NHEADS * 4);
  float*    denom = (float*)   wsalloc(ws, off, (size_t)NN * NHEADS * 4);
  float*    ebuf  = (float*)   wsalloc(ws, off, (size_t)ETOT * NHEADS * 4);

  const dim3 wmmaBlk(32, 1, 1);

  // ---- layer 0: x[N,128] @ W1 -> h0; attention; agg -> h1; +b1; elu ----
  gemm_f32_wmma<DINF, HIDF><<<dim3(NN / 16, HIDF / 16), wmmaBlk, 0, stream>>>(x, W1, h0);
  run_attention(ei, h0, as1, ad1, b1, h1, als, ald, menc, denom, ebuf,
                NHEADS, DHH, /*elu=*/1, stream);

  // ---- hidden layers: h1 -> h0 (gemm), agg back into h1 ----
  for (int i = 0; i < NLHID; ++i) {
    gemm_f32_wmma<HIDF, HIDF><<<dim3(NN / 16, HIDF / 16), wmmaBlk, 0, stream>>>(
        h1, Wh + (size_t)i * HIDF * HIDF, h0);
    // h1 is free after the gemm (stream-ordered); reuse as aggregation target
    run_attention(ei, h0, ash + (size_t)i * NHEADS * DHH,
                  adh + (size_t)i * NHEADS * DHH, bh + (size_t)i * HIDF, h1,
                  als, ald, menc, denom, ebuf, NHEADS, DHH, /*elu=*/1, stream);
  }

  // ---- output layer: h1[N,128] @ W2[128,16] -> h0; heads=1; agg -> d_out ----
  gemm_f32_wmma<HIDF, DOUTF><<<dim3(NN / 16, DOUTF / 16), wmmaBlk, 0, stream>>>(h1, W2, h0);
  run_attention(ei, h0, as2, ad2, b2, out, als, ald, menc, denom, ebuf,
                /*heads=*/1, /*ch=*/DOUTF, /*elu=*/0, stream);
}